// SPIN_48095043781088
// MI455X (gfx1250) — compile-verified
//
#include <hip/hip_runtime.h>

// Problem constants (from reference setup_inputs):
//   pixel_features (2, 64, 320, 320) f32, stoken = 16x16
#define BB   2
#define CC   64
#define HH   320
#define WW   320
#define SHH  16
#define SWW  16
#define NSH  20            // HH/SHH
#define NSW  20            // WW/SWW
#define NSP  400           // superpixels
#define NPIX 102400        // HH*WW

typedef __attribute__((ext_vector_type(2))) float v2f;
typedef __attribute__((ext_vector_type(4))) float v4f;
typedef __attribute__((ext_vector_type(8))) float v8f;

// ---------------------------------------------------------------- centroids
// one wave per (b, s, c): mean over the 16x16 block.
__global__ __launch_bounds__(256) void k_centroid(const float* __restrict__ pf,
                                                  float* __restrict__ spix) {
  int wid  = blockIdx.x * 8 + (threadIdx.x >> 5);
  int lane = threadIdx.x & 31;
  int b = wid / (NSP * CC);
  int s = (wid / CC) % NSP;
  int c = wid % CC;
  int r0 = (s / NSW) * SHH, c0 = (s % NSW) * SWW;
  const float* src = pf + (size_t)(b * CC + c) * NPIX;
  float sum = 0.f;
  for (int i = lane; i < SHH * SWW; i += 32) {
    int p = (r0 + (i >> 4)) * WW + c0 + (i & 15);
    sum += src[p];
  }
  for (int off = 16; off; off >>= 1) sum += __shfl_xor(sum, off, 32);
  if (lane == 0) spix[(b * NSP + s) * CC + c] = sum * (1.0f / (SHH * SWW));
}

// ---------------------------------------------------------------- distances
// One block per cell (b, s). GEMM (256 px x 64 ch) x (64 ch x 16 cand) via
// V_WMMA_F32_16X16X4_F32; then d = |x|^2 - 2 x.c + |c|^2, softmax over the
// 9 valid candidates, stored compact: aff[b][p][k], k in 0..8.
__global__ __launch_bounds__(256) void k_dist(const float* __restrict__ pf,
                                              const float* __restrict__ spix,
                                              float* __restrict__ aff) {
  __shared__ float centT[CC * 16];   // [channel][cand], cand padded to 16
  __shared__ float cnorm[16];
  int tid  = threadIdx.x;
  int cell = blockIdx.x;
  int b = cell / NSP, s = cell % NSP;
  int sr = s / NSW, sc = s % NSW;
  int r0 = sr * SHH, c0 = sc * SWW;

  for (int i = tid; i < CC * 16; i += 256) centT[i] = 0.f;
  __syncthreads();
  for (int t = tid; t < 9 * CC; t += 256) {
    int n = t / CC, c = t % CC;
    int rr = sr + n / 3 - 1, c2 = sc + n % 3 - 1;
    if (rr >= 0 && rr < NSH && c2 >= 0 && c2 < NSW)
      centT[c * 16 + n] = spix[((b * NSP) + rr * NSW + c2) * CC + c];
  }
  __syncthreads();
  if (tid < 16) {
    float sn = 0.f;
    for (int c = 0; c < CC; ++c) { float v = centT[c * 16 + tid]; sn += v * v; }
    cnorm[tid] = sn;
  }
  __syncthreads();

  int wave = tid >> 5, lane = tid & 31;
  int half = lane >> 4, m = lane & 15;
  int n = m;                                  // candidate column this lane owns
  int vdr = n / 3 - 1, vdc = n % 3 - 1;
  bool nvalid = (n < 9) && (sr + vdr >= 0) && (sr + vdr < NSH) &&
                (sc + vdc >= 0) && (sc + vdc < NSW);
  float cn = cnorm[n];

  for (int tt = 0; tt < 2; ++tt) {
    int t = wave * 2 + tt;                    // cell row handled by this tile
    int prow = (r0 + t) * WW + c0;
    if (tt == 0) {
      // prefetch the first A-rows of the next tile (emits global_prefetch_b8)
      __builtin_prefetch(&pf[(size_t)(b * CC + 2 * half) * NPIX + prow + WW + m]);
    }
    v8f acc = {};
    float psq = 0.f;
    for (int kk = 0; kk < CC / 4; ++kk) {
      int kb = kk * 4 + 2 * half;             // A lanes 0-15: K,K+1 ; 16-31: K+2,K+3
      v2f a, bm;
      a.x = pf[(size_t)(b * CC + kb    ) * NPIX + prow + m];
      a.y = pf[(size_t)(b * CC + kb + 1) * NPIX + prow + m];
      psq += a.x * a.x + a.y * a.y;
      bm.x = centT[(kb    ) * 16 + m];
      bm.y = centT[(kb + 1) * 16 + m];
      acc = __builtin_amdgcn_wmma_f32_16x16x4_f32(false, a, false, bm,
                                                  (short)0, acc, false, false);
    }
    float psqF = psq + __shfl_xor(psq, 16, 32);   // full |x|^2 of pixel col m

    float nd[8], av[8];
    for (int v = 0; v < 8; ++v) {
      float pm = __shfl(psqF, v + 8 * half, 32);  // pixel M = v + 8*half
      float d  = pm - 2.0f * acc[v] + cn;
      nd[v] = nvalid ? -d : -__builtin_inff();
    }
    for (int v = 0; v < 8; ++v) {
      float mx = nd[v];
      for (int off = 8; off; off >>= 1) mx = fmaxf(mx, __shfl_xor(mx, off, 32));
      float e = nvalid ? __expf(nd[v] - mx) : 0.f;
      float ss = e;
      for (int off = 8; off; off >>= 1) ss += __shfl_xor(ss, off, 32);
      av[v] = e / ss;
    }
    if (n < 9) {
      for (int v = 0; v < 8; ++v) {
        int p = prow + v + 8 * half;
        aff[((size_t)b * NPIX + p) * 9 + n] = av[v];
      }
    }
  }
}

// ---------------------------------------------------------------- per-cell partials
// Each cell reads its own 256 pixels ONCE and emits, for each of its 9
// candidate slots k:  partial[cell][k][c] = sum_p aff[p][k] * pf[c][p]
// and pden[cell][k] = sum_p aff[p][k].  (invalid slots hold aff==0)
__global__ __launch_bounds__(256) void k_partial(const float* __restrict__ pf,
                                                 const float* __restrict__ aff,
                                                 float* __restrict__ partial,
                                                 float* __restrict__ pden) {
  __shared__ float affs[256 * 9];
  int tid  = threadIdx.x;
  int cell = blockIdx.x;                  // b*NSP + s
  int b = cell / NSP, s = cell % NSP;
  int sr = s / NSW, sc = s % NSW;
  int r0 = sr * SHH, c0 = sc * SWW;

  for (int idx = tid; idx < 256 * 9; idx += 256) {
    int i = idx / 9, k = idx % 9;
    int p = (r0 + (i >> 4)) * WW + c0 + (i & 15);
    affs[idx] = aff[((size_t)b * NPIX + p) * 9 + k];
  }
  __syncthreads();

  int wave = tid >> 5, lane = tid & 31;
  if (wave == 0) {
    for (int k = 0; k < 9; ++k) {
      float dsum = 0.f;
      for (int i = lane; i < 256; i += 32) dsum += affs[i * 9 + k];
      for (int off = 16; off; off >>= 1) dsum += __shfl_xor(dsum, off, 32);
      if (lane == 0) pden[cell * 9 + k] = dsum;
    }
  }
  for (int c8 = 0; c8 < 8; ++c8) {
    int c = wave * 8 + c8;
    const float* src = pf + (size_t)(b * CC + c) * NPIX;
    float acc[9];
    for (int k = 0; k < 9; ++k) acc[k] = 0.f;
    for (int i = lane; i < 256; i += 32) {
      int p = (r0 + (i >> 4)) * WW + c0 + (i & 15);
      float x = src[p];
      for (int k = 0; k < 9; ++k) acc[k] += affs[i * 9 + k] * x;
    }
    for (int k = 0; k < 9; ++k) {
      float v = acc[k];
      for (int off = 16; off; off >>= 1) v += __shfl_xor(v, off, 32);
      if (lane == 0) partial[((size_t)cell * 9 + k) * CC + c] = v;
    }
  }
}

// ---------------------------------------------------------------- centroid gather
// spix[b][s][c] = (sum of 9 neighbor-cell partials) / (den + 1e-16)
__global__ __launch_bounds__(256) void k_spix(const float* __restrict__ partial,
                                              const float* __restrict__ pden,
                                              float* __restrict__ spix) {
  int gid = blockIdx.x * 256 + threadIdx.x;   // over B*NSP*CC
  int c = gid % CC;
  int s = (gid / CC) % NSP;
  int b = gid / (CC * NSP);
  int sr = s / NSW, sc = s % NSW;
  float num = 0.f, den = 0.f;
  for (int ci = 0; ci < 9; ++ci) {
    int dr = ci / 3 - 1, dc = ci % 3 - 1;
    int rr = sr + dr, c2 = sc + dc;
    if (rr < 0 || rr >= NSH || c2 < 0 || c2 >= NSW) continue;
    int ncell = b * NSP + rr * NSW + c2;
    int k = (1 - dr) * 3 + (1 - dc);          // slot under which that cell sees s
    num += partial[((size_t)ncell * 9 + k) * CC + c];
    den += pden[ncell * 9 + k];
  }
  spix[(b * NSP + s) * CC + c] = num / (den + 1e-16f);
}

// ---------------------------------------------------------------- dense expand
// Streaming float4 NON-TEMPORAL writes of the full (B, ns, N) output (328 MB
// > 192 MB L2 -> keep it out of L2 so pf/aff stay resident).
__global__ __launch_bounds__(256) void k_expand(const float* __restrict__ aff,
                                                float* __restrict__ out) {
  int gid = blockIdx.x * 256 + threadIdx.x;       // over B*NSP*NPIX/4
  const int PQ = NPIX / 4;
  int p4 = (gid % PQ) * 4;
  int s  = (gid / PQ) % NSP;
  int b  = gid / (PQ * NSP);
  int sr = s / NSW, sc = s % NSW;
  v4f r;
  for (int e = 0; e < 4; ++e) {
    int p = p4 + e;
    int row = p / WW, col = p % WW;
    int dr = sr - row / SHH, dc = sc - col / SWW;
    float v = 0.f;
    if (dr >= -1 && dr <= 1 && dc >= -1 && dc <= 1) {
      int k = (dr + 1) * 3 + (dc + 1);
      v = aff[((size_t)b * NPIX + p) * 9 + k];
    }
    r[e] = v;
  }
  __builtin_nontemporal_store(r, (v4f*)(out + (size_t)(b * NSP + s) * NPIX + p4));
}

extern "C" void kernel_launch(void* const* d_in, const int* in_sizes, int n_in,
                              void* d_out, int out_size, void* d_ws, size_t ws_size,
                              hipStream_t stream) {
  (void)in_sizes; (void)n_in; (void)out_size; (void)ws_size;
  const float* pf = (const float*)d_in[0];        // (B,C,H,W) f32; stoken_h/w hardcoded (16)
  float* out     = (float*)d_out;                 // (B, ns, N) f32 dense
  float* spix    = (float*)d_ws;                              // B*NSP*CC
  float* aff     = spix    + (size_t)BB * NSP * CC;           // B*NPIX*9
  float* partial = aff     + (size_t)BB * NPIX * 9;           // B*NSP*9*CC
  float* pden    = partial + (size_t)BB * NSP * 9 * CC;       // B*NSP*9

  k_centroid<<<BB * NSP * CC / 8, 256, 0, stream>>>(pf, spix);
  k_dist    <<<BB * NSP,          256, 0, stream>>>(pf, spix, aff);   // iter 0
  k_partial <<<BB * NSP,          256, 0, stream>>>(pf, aff, partial, pden);
  k_spix    <<<BB * NSP * CC / 256, 256, 0, stream>>>(partial, pden, spix);
  k_dist    <<<BB * NSP,          256, 0, stream>>>(pf, spix, aff);   // iter 1
  k_expand  <<<(BB * NSP * NPIX / 4) / 256, 256, 0, stream>>>(aff, out);
}